// PaiNNBlock_62663572848823
// MI455X (gfx1250) — compile-verified
//
#include <hip/hip_runtime.h>
#include <hip/hip_bf16.h>

typedef __attribute__((ext_vector_type(2))) float v2f;
typedef __attribute__((ext_vector_type(8))) float v8f;

#define N_NODES 10000
#define N_EDGES 256000

__device__ __forceinline__ float silu_f(float x) {
    return x / (1.0f + __expf(-x));
}

// ---------------------------------------------------------------------------
// Generic fp32 GEMM: out[M,N] = maybe_silu(A[M,K] @ W[K,N] + bias[N])
// One wave computes a 16x32 strip (2 n-tiles sharing the A fragment).
// V_WMMA_F32_16X16X4_F32 fragment layouts (ISA 7.12.2):
//   A 16x4 : lane%16 = row M; v.x = K(k0+2*half), v.y = K(k0+2*half+1)
//   B 4x16 : lane%16 = col N; v.x = row K(k0+2*half), v.y = K(k0+2*half+1)
//   C 16x16: lane%16 = col N; VGPR g = row (g + 8*half)
// M%16==0, N%32==0, K%4==0 (true for all calls here).
// ---------------------------------------------------------------------------
__global__ __launch_bounds__(256) void gemm_f32_wmma_kernel(
    const float* __restrict__ A, const float* __restrict__ W,
    const float* __restrict__ bias, float* __restrict__ out,
    int M, int K, int N, int apply_silu)
{
    const int wave = threadIdx.x >> 5;
    const int lane = threadIdx.x & 31;
    const int l    = lane & 15;
    const int half = lane >> 4;

    const int nstrips = N >> 5;
    const int strip   = blockIdx.x * 8 + wave;
    const int total   = (M >> 4) * nstrips;
    if (strip >= total) return;           // wave-uniform: EXEC stays all-1s

    const int mt = strip / nstrips;
    const int ns = strip % nstrips;
    const int m0 = mt << 4;
    const int n0 = ns << 5;

    v8f acc0, acc1;
    const float b0 = bias ? bias[n0 + l]      : 0.0f;
    const float b1 = bias ? bias[n0 + 16 + l] : 0.0f;
#pragma unroll
    for (int g = 0; g < 8; ++g) { acc0[g] = b0; acc1[g] = b1; }

    const float* Arow = A + (m0 + l) * K;
    for (int k0 = 0; k0 < K; k0 += 4) {
        v2f a, bA, bB;
        a.x = Arow[k0 + 2 * half];
        a.y = Arow[k0 + 2 * half + 1];
        const float* Wr0 = W + (k0 + 2 * half) * N + n0;
        const float* Wr1 = W + (k0 + 2 * half + 1) * N + n0;
        bA.x = Wr0[l];        bA.y = Wr1[l];
        bB.x = Wr0[16 + l];   bB.y = Wr1[16 + l];
        acc0 = __builtin_amdgcn_wmma_f32_16x16x4_f32(false, a, false, bA,
                                                     (short)0, acc0, false, false);
        acc1 = __builtin_amdgcn_wmma_f32_16x16x4_f32(false, a, false, bB,
                                                     (short)0, acc1, false, false);
    }

#pragma unroll
    for (int g = 0; g < 8; ++g) {
        const int row = m0 + g + 8 * half;
        float v0 = acc0[g], v1 = acc1[g];
        if (apply_silu) { v0 = silu_f(v0); v1 = silu_f(v1); }
        out[row * N + n0 + l]      = v0;
        out[row * N + n0 + 16 + l] = v1;
    }
}

// ---------------------------------------------------------------------------
// Fused edge kernel: per block = 16 edges, 8 waves.
//   filters = (silu(rbf@Wf1+bf1) @ Wf2 + bf2) * cutoff   -- all via WMMA
//   then gather X[src], mu[src] and atomically scatter messages to
//   q_acc[tgt] and mu_acc[tgt].
// Wave w owns channel columns [16w, 16w+16): its 3 layer-2 tiles are exactly
// filter_q / filter_r / filter_mu for those columns, so C-fragment element
// (g, lane) maps to (edge = g+8*half, channel = 16w + lane%16).
// ---------------------------------------------------------------------------
__global__ __launch_bounds__(256) void edge_msg_kernel(
    const float* __restrict__ X,        // [N,384] node inter-MLP output
    const float* __restrict__ mu,       // [N,3,128] original mu
    const int*   __restrict__ edge_index, // [2,E] (row0=target, row1=source)
    const float* __restrict__ rbf,      // [E,20]
    const float* __restrict__ uv,       // [E,3]
    const float* __restrict__ cut,      // [E]
    const float* __restrict__ Wf1,      // [20,128]
    const float* __restrict__ bf1,      // [128]
    const float* __restrict__ Wf2,      // [128,384]
    const float* __restrict__ bf2,      // [384]
    float* __restrict__ q_acc,          // [N,128]  (pre-init with q)
    float* __restrict__ mu_acc,         // [N,3,128] (pre-init with mu)
    int E)
{
    __shared__ float s_rbf[16][20];
    __shared__ float s_h[16][128];
    __shared__ int   s_src[16];
    __shared__ int   s_tgt[16];
    __shared__ float s_uv[16][3];
    __shared__ float s_cut[16];

    const int tid = threadIdx.x;
    const int e0  = blockIdx.x * 16;

    for (int i = tid; i < 16 * 20; i += 256)
        s_rbf[i / 20][i % 20] = rbf[e0 * 20 + i];
    if (tid < 16) {
        s_tgt[tid] = edge_index[e0 + tid];
        s_src[tid] = edge_index[E + e0 + tid];
        s_cut[tid] = cut[e0 + tid];
    }
    if (tid < 48) s_uv[tid / 3][tid % 3] = uv[e0 * 3 + tid];
    __syncthreads();

    const int wave = tid >> 5;
    const int lane = tid & 31;
    const int l    = lane & 15;
    const int half = lane >> 4;

    // ---- layer 1: h = silu(rbf @ Wf1 + bf1); wave w -> cols [16w,16w+16)
    {
        const int n0 = wave * 16;
        v8f acc;
        const float bv = bf1[n0 + l];
#pragma unroll
        for (int g = 0; g < 8; ++g) acc[g] = bv;
#pragma unroll
        for (int k0 = 0; k0 < 20; k0 += 4) {
            v2f a, b;
            a.x = s_rbf[l][k0 + 2 * half];
            a.y = s_rbf[l][k0 + 2 * half + 1];
            b.x = Wf1[(k0 + 2 * half) * 128 + n0 + l];
            b.y = Wf1[(k0 + 2 * half + 1) * 128 + n0 + l];
            acc = __builtin_amdgcn_wmma_f32_16x16x4_f32(false, a, false, b,
                                                        (short)0, acc, false, false);
        }
#pragma unroll
        for (int g = 0; g < 8; ++g)
            s_h[g + 8 * half][n0 + l] = silu_f(acc[g]);
    }
    __syncthreads();

    // ---- layer 2: filters = h @ Wf2 + bf2; wave w -> tiles {w, w+8, w+16}
    v8f facc[3];
#pragma unroll
    for (int t = 0; t < 3; ++t) {
        const int c = wave + 8 * t;
        const float bv = bf2[c * 16 + l];
#pragma unroll
        for (int g = 0; g < 8; ++g) facc[t][g] = bv;
    }
    for (int k0 = 0; k0 < 128; k0 += 4) {
        v2f a;
        a.x = s_h[l][k0 + 2 * half];
        a.y = s_h[l][k0 + 2 * half + 1];
#pragma unroll
        for (int t = 0; t < 3; ++t) {
            const int c = wave + 8 * t;
            v2f b;
            b.x = Wf2[(k0 + 2 * half) * 384 + c * 16 + l];
            b.y = Wf2[(k0 + 2 * half + 1) * 384 + c * 16 + l];
            facc[t] = __builtin_amdgcn_wmma_f32_16x16x4_f32(false, a, false, b,
                                                            (short)0, facc[t], false, false);
        }
    }

    // ---- cutoff scaling (per edge = per fragment row)
#pragma unroll
    for (int t = 0; t < 3; ++t)
#pragma unroll
        for (int g = 0; g < 8; ++g)
            facc[t][g] *= s_cut[g + 8 * half];

    // ---- gather / multiply / scatter messages
    const int col = wave * 16 + l;   // channel in [0,128)
#pragma unroll
    for (int g = 0; g < 8; ++g) {
        const int er  = g + 8 * half;     // edge within block
        const int src = s_src[er];
        const int tgt = s_tgt[er];
        const float fq = facc[0][g];
        const float fr = facc[1][g];
        const float fm = facc[2][g];
        const float* xb = X + src * 384;
        const float  xq = xb[col];
        const float  xr = xb[128 + col];
        const float  xm = xb[256 + col];
        atomicAdd(&q_acc[tgt * 128 + col], xq * fq);
        const float xrf = xr * fr;
        const float xmf = xm * fm;
        const float* mb = mu + src * 384;
        float* ab = mu_acc + tgt * 384;
#pragma unroll
        for (int d = 0; d < 3; ++d)
            atomicAdd(&ab[d * 128 + col], s_uv[er][d] * xrf + mb[d * 128 + col] * xmf);
    }
}

// ---------------------------------------------------------------------------
// mix prep: mu_v_norm, inner product, scalar_input = [q_upd | norm]
// mucat layout: row r = n*3 + d, cols 0..127 = mu_v, 128..255 = mu_w
// ---------------------------------------------------------------------------
__global__ __launch_bounds__(256) void mix_prep_kernel(
    const float* __restrict__ mucat, const float* __restrict__ q_acc,
    float* __restrict__ sc_in, float* __restrict__ inner, int N)
{
    const int idx = blockIdx.x * blockDim.x + threadIdx.x;
    if (idx >= N * 128) return;
    const int n = idx >> 7;
    const int h = idx & 127;
    const float* base = mucat + n * 3 * 256;
    const float v0 = base[h],       v1 = base[256 + h],       v2 = base[512 + h];
    const float w0 = base[128 + h], w1 = base[384 + h],       w2 = base[640 + h];
    sc_in[n * 256 + h]       = q_acc[idx];
    sc_in[n * 256 + 128 + h] = sqrtf(v0 * v0 + v1 * v1 + v2 * v2 + 1e-8f);
    inner[idx]               = v0 * w0 + v1 * w1 + v2 * w2;
}

// ---------------------------------------------------------------------------
// finalize: q_out = q_upd + dq + dqmu*inner ; mu_out = mu_upd + mu_w*dmu_scale
// d_out = [q_out (N*128) | mu_out (N*3*128)]
// ---------------------------------------------------------------------------
__global__ __launch_bounds__(256) void finalize_kernel(
    const float* __restrict__ q_acc, const float* __restrict__ mu_acc,
    const float* __restrict__ delta, const float* __restrict__ mucat,
    const float* __restrict__ inner, float* __restrict__ out, int N)
{
    const int idx = blockIdx.x * blockDim.x + threadIdx.x;
    if (idx >= N * 128) return;
    const int n = idx >> 7;
    const int h = idx & 127;
    const float* db = delta + n * 384;
    const float dq     = db[h];
    const float dscale = db[128 + h];
    const float dqmu   = db[256 + h];
    out[idx] = q_acc[idx] + dq + dqmu * inner[idx];
    const float* base = mucat + n * 3 * 256;
    const float* mab  = mu_acc + n * 384;
    float* ob = out + N * 128 + n * 384;
#pragma unroll
    for (int d = 0; d < 3; ++d)
        ob[d * 128 + h] = mab[d * 128 + h] + base[d * 256 + 128 + h] * dscale;
}

extern "C" void kernel_launch(void* const* d_in, const int* in_sizes, int n_in,
                              void* d_out, int out_size, void* d_ws, size_t ws_size,
                              hipStream_t stream) {
    const float* q        = (const float*)d_in[0];
    const float* mu       = (const float*)d_in[1];
    const int*   eidx     = (const int*)d_in[2];
    const float* rbf      = (const float*)d_in[3];
    const float* uv       = (const float*)d_in[4];
    const float* cut      = (const float*)d_in[5];
    const float* W_inter1 = (const float*)d_in[6];
    const float* b_inter1 = (const float*)d_in[7];
    const float* W_inter2 = (const float*)d_in[8];
    const float* b_inter2 = (const float*)d_in[9];
    const float* W_filt1  = (const float*)d_in[10];
    const float* b_filt1  = (const float*)d_in[11];
    const float* W_filt2  = (const float*)d_in[12];
    const float* b_filt2  = (const float*)d_in[13];
    const float* W_vec    = (const float*)d_in[14];
    const float* W_mix1   = (const float*)d_in[15];
    const float* b_mix1   = (const float*)d_in[16];
    const float* W_mix2   = (const float*)d_in[17];
    const float* b_mix2   = (const float*)d_in[18];

    const int N = N_NODES;
    const int E = N_EDGES;

    float* ws     = (float*)d_ws;
    float* q_acc  = ws;                 // N*128   = 1,280,000
    float* mu_acc = ws + 1280000;       // N*384   = 3,840,000
    float* H1     = ws + 5120000;       // N*384
    float* X      = ws + 8960000;       // N*384
    float* mucat  = ws + 12800000;      // 3N*256  = 7,680,000
    float* sc_in  = ws + 20480000;      // N*256   = 2,560,000
    float* innerb = ws + 23040000;      // N*128   = 1,280,000
    float* Hm     = H1;                 // reuse (H1 dead after X)
    float* delta  = X;                  // reuse (X dead after edge kernel)

    // accumulators start at q / mu; edge messages atomicAdd on top
    hipMemcpyAsync(q_acc,  q,  (size_t)N * 128 * sizeof(float), hipMemcpyDeviceToDevice, stream);
    hipMemcpyAsync(mu_acc, mu, (size_t)N * 384 * sizeof(float), hipMemcpyDeviceToDevice, stream);

    auto gemm = [&](const float* A, const float* W, const float* b, float* O,
                    int M, int K, int Nc, int silu) {
        const int strips = (M / 16) * (Nc / 32);
        const int blocks = (strips + 7) / 8;
        gemm_f32_wmma_kernel<<<blocks, 256, 0, stream>>>(A, W, b, O, M, K, Nc, silu);
    };

    // node inter MLP
    gemm(q,  W_inter1, b_inter1, H1, N, 128, 384, 1);
    gemm(H1, W_inter2, b_inter2, X,  N, 384, 384, 0);

    // fused edge filter MLP + message gather/scatter
    edge_msg_kernel<<<E / 16, 256, 0, stream>>>(X, mu, eidx, rbf, uv, cut,
                                                W_filt1, b_filt1, W_filt2, b_filt2,
                                                q_acc, mu_acc, E);

    // mixing
    gemm(mu_acc, W_vec, nullptr, mucat, 3 * N, 128, 256, 0);
    mix_prep_kernel<<<(N * 128 + 255) / 256, 256, 0, stream>>>(mucat, q_acc, sc_in, innerb, N);
    gemm(sc_in, W_mix1, b_mix1, Hm,    N, 256, 384, 1);
    gemm(Hm,    W_mix2, b_mix2, delta, N, 384, 384, 0);

    finalize_kernel<<<(N * 128 + 255) / 256, 256, 0, stream>>>(
        q_acc, mu_acc, delta, mucat, innerb, (float*)d_out, N);
}